// AttentionBasedConvLSTM_7310034338021
// MI455X (gfx1250) — compile-verified
//
#include <hip/hip_runtime.h>

// ---------------------------------------------------------------------------
// Attention-based 2-layer ConvLSTM for MI455X (gfx1250, wave32).
// Convs = implicit GEMM on V_WMMA_F32_16X16X4_F32; weight panels streamed
// into double-buffered LDS by the Tensor Data Mover (tensor_load_to_lds),
// synchronized with s_wait_tensorcnt + one workgroup barrier per K-step.
// ---------------------------------------------------------------------------

typedef __attribute__((ext_vector_type(2))) float v2f;
typedef __attribute__((ext_vector_type(8))) float v8f;
typedef __attribute__((ext_vector_type(4))) unsigned int v4u;
typedef __attribute__((ext_vector_type(8))) int v8i;
typedef __attribute__((ext_vector_type(4))) int v4i;

#define BATCH 8
#define TSTEPS 16
#define HID 64
#define HW 4096           // 64*64
#define GATES 256         // 4*HID

// K layout: k = rs*CINPAD + ci  (rs = r*3+s tap index, ci = input channel)
// layer0: Cin=67 -> CINPAD=72,  KPAD=9*72 =648
// layer1: Cin=128-> CINPAD=128, KPAD=9*128=1152

// workspace layout (float offsets)
#define XIN_OFF   0                        // [B][T][3][HW]      = 1572864
#define H0_OFF    1572864                  // [B][64][HW]        = 2097152
#define C0_OFF    3670016                  // [B][64][HW]        = 2097152
#define GATE_OFF  5767168                  // [B][256][HW]       = 8388608
#define WT0_OFF   14155776                 // 648*256            = 165888
#define WT1_OFF   14321664                 // 1152*256           = 294912
// total 14616576 floats ~= 55.8 MB

// d_out layout (float offsets)
#define OUT1_OFF  0                        // [B][T][64][HW]     = 33554432
#define H1_OFF    33554432                 // [B][64][HW]
#define C1_OFF    35651584                 // [B][64][HW]

__device__ __forceinline__ float sigmoidf(float x) {
    return 1.0f / (1.0f + __expf(-x));
}

// ---------------------------------------------------------------------------
// TDM: DMA one contiguous 8 KB weight panel (2048 fp32) global -> LDS.
// D# group0: count=1, lds_addr, 57b global addr, type=2 (bits 127:126).
// D# group1: data_size=4B, 1-D tile: tensor_dim0=tile_dim0=stride=2048.
// Groups 2/3 + trailing group: zero (tile dims 3/4 unused, no iterate).
// ---------------------------------------------------------------------------
__device__ __forceinline__ void tdm_load_panel(unsigned int lds_addr,
                                               const float* gptr) {
    unsigned long long ga = (unsigned long long)(uintptr_t)gptr;
    v4u g0 = {1u,                                   // count = 1 (valid D#)
              lds_addr,                             // LDS byte address
              (unsigned int)ga,                     // global_addr[31:0]
              (unsigned int)(ga >> 32) | 0x80000000u};  // addr[56:32] | type=2
    v8i g1 = {0x00020000,        // data_size = 2 (4 bytes/elem)
              0x08000000,        // tensor_dim0 = 2048 (bits 63:48)
              0,                 // tensor_dim0 hi / tensor_dim1 lo
              0x08000000,        // tile_dim0 = 2048 (bits 127:112)
              0,                 // tile_dim1 = 0 (1-D), tile_dim2 = 0
              2048,              // tensor_dim0_stride[31:0]
              0, 0};
    v4i z4 = {0, 0, 0, 0};
    v8i z8 = {0, 0, 0, 0, 0, 0, 0, 0};
    __builtin_amdgcn_tensor_load_to_lds(g0, g1, z4, z4, z8, 0);
}

// ---------------------------------------------------------------------------
// Weight pack: w[256][Cin][3][3] -> Wt, K = rs*CINPAD+ci (zero-pad ci>=Cin),
// pair-interleaved for ds_load_b64 A fetch: Wt[(k>>1)*512 + m*2 + (k&1)].
// grid = KPAD blocks x 256 threads
// ---------------------------------------------------------------------------
template <int CIN, int CINPAD>
__global__ __launch_bounds__(256) void wt_pack_kernel(
    const float* __restrict__ w, float* __restrict__ wt) {
    const int e = blockIdx.x * 256 + threadIdx.x;
    const int m = (e >> 1) & 255;
    const int k = ((e >> 9) << 1) | (e & 1);
    const int rs = k / CINPAD;
    const int ci = k - rs * CINPAD;
    wt[e] = (ci < CIN) ? w[((size_t)m * CIN + ci) * 9 + rs] : 0.0f;
}

// ---------------------------------------------------------------------------
// SE-style channel attention + gating:  xin = x * sigmoid(fc(avg)+fc(max))
// grid = B*T blocks x 256 threads
// ---------------------------------------------------------------------------
__global__ __launch_bounds__(256) void attention_kernel(
    const float* __restrict__ x, const float* __restrict__ w1,
    const float* __restrict__ w2, float* __restrict__ xin) {
    __shared__ float rs_[256];
    __shared__ float rm_[256];
    __shared__ float s_avg[3];
    __shared__ float s_max[3];
    const int tid = threadIdx.x;
    const size_t base = (size_t)blockIdx.x * 3 * HW;   // blockIdx.x = b*T+t

    for (int c = 0; c < 3; ++c) {
        float s = 0.0f, m = -3.4e38f;
        const float* __restrict__ xc = x + base + (size_t)c * HW;
        for (int i = tid; i < HW; i += 256) {
            float v = xc[i];
            s += v;
            m = fmaxf(m, v);
        }
        rs_[tid] = s; rm_[tid] = m;
        __syncthreads();
        for (int off = 128; off > 0; off >>= 1) {
            if (tid < off) {
                rs_[tid] += rs_[tid + off];
                rm_[tid] = fmaxf(rm_[tid], rm_[tid + off]);
            }
            __syncthreads();
        }
        if (tid == 0) { s_avg[c] = rs_[0] * (1.0f / HW); s_max[c] = rm_[0]; }
        __syncthreads();
    }

    const float w10 = w1[0], w11 = w1[1], w12 = w1[2];
    float ra = fmaxf(0.0f, w10 * s_avg[0] + w11 * s_avg[1] + w12 * s_avg[2]);
    float rx = fmaxf(0.0f, w10 * s_max[0] + w11 * s_max[1] + w12 * s_max[2]);
    for (int c = 0; c < 3; ++c) {
        float sc = sigmoidf(w2[c] * ra + w2[c] * rx);
        const float* __restrict__ xc = x + base + (size_t)c * HW;
        float* __restrict__ oc = xin + base + (size_t)c * HW;
        for (int i = tid; i < HW; i += 256) oc[i] = xc[i] * sc;
    }
}

// ---------------------------------------------------------------------------
// Implicit-GEMM 3x3 SAME conv -> gate pre-activations via fp32 WMMA 16x16x4.
//   M=256 gate channels, N=128 pixels/block (16 per wave), K=9*CINPAD.
//   A panels (8 K-rows, 8 KB) TDM-DMA'd into double-buffered LDS; per
//   iteration the tap (r,s) is uniform, so B im2col is 4 channel loads/lane.
// grid = dim3(32, B) x 256 threads (8 wave32)
// ---------------------------------------------------------------------------
template <int CINX, int CINTOT, int CINPAD, int KPAD>
__global__ __launch_bounds__(256) void conv_gates_kernel(
    const float* __restrict__ xp, long bsx,       // x-part base, batch stride
    const float* __restrict__ hp,                 // h-part [B][64][HW]
    const float* __restrict__ Wt,                 // packed [KPAD/2][256][2]
    float* __restrict__ gates) {                  // [B][256][HW]
    __shared__ __align__(16) float ldsA[2][2048];  // 2 x 8 KB weight panels
    const int tid  = threadIdx.x;
    const int lane = tid & 31;
    const int hi   = lane >> 4;
    const int lm   = lane & 15;
    const int b    = blockIdx.y;
    const int Nbase = blockIdx.x * 128 + (tid >> 5) * 16;
    const int n = Nbase + lm;
    const int y = n >> 6;
    const int x = n & 63;

    // wave 0 drives the TDM; force a scalar (wave-uniform) predicate
    const bool lead = __builtin_amdgcn_readfirstlane((int)(tid < 32)) != 0;
    const unsigned int lds0 = (unsigned int)(uintptr_t)&ldsA[0][0];

    const v8f vzero = {0.f, 0.f, 0.f, 0.f, 0.f, 0.f, 0.f, 0.f};
    v8f acc[16];
#pragma unroll
    for (int i = 0; i < 16; ++i) acc[i] = vzero;

    const float* __restrict__ xb = xp + (size_t)b * (size_t)bsx;
    const float* __restrict__ hb = hp + (size_t)b * (size_t)HID * HW;

    if (lead) tdm_load_panel(lds0, Wt);           // prologue: panel for k0=0

    int p = 0;
    for (int k0 = 0; k0 < KPAD; k0 += 8) {
        if (lead) __builtin_amdgcn_s_wait_tensorcnt(0);  // panel p landed
        __syncthreads();   // publish panel p; retires all reads of panel p^1
        if (lead && (k0 + 8 < KPAD))
            tdm_load_panel(lds0 + (unsigned int)((p ^ 1) * 8192),
                           Wt + (size_t)(k0 + 8) * 256);

        // uniform tap for this K block
        const int rs = k0 / CINPAD;
        const int cibase = k0 - rs * CINPAD;
        const int yy = y + (rs / 3) - 1;
        const int xx = x + (rs % 3) - 1;
        const bool inb = ((unsigned)yy < 64u) && ((unsigned)xx < 64u);
        const int pix = (yy << 6) + xx;

        // B operands: 2 sub-chunks of K=4; VGPR j / lane-group hi -> K=2*hi+j
        v2f bv[2];
#pragma unroll
        for (int q = 0; q < 2; ++q) {
#pragma unroll
            for (int j = 0; j < 2; ++j) {
                const int ci = cibase + 4 * q + 2 * hi + j;
                float vb = 0.0f;
                if (inb && (CINPAD == CINTOT || ci < CINTOT)) {
                    vb = (ci < CINX) ? xb[(size_t)ci * HW + pix]
                                     : hb[(size_t)(ci - CINX) * HW + pix];
                }
                bv[q][j] = vb;
            }
        }

        const float* __restrict__ panel = &ldsA[p][0];
#pragma unroll
        for (int q = 0; q < 2; ++q) {
#pragma unroll
            for (int mt = 0; mt < 16; ++mt) {
                // pair-interleaved panel: element(k,m) at (k>>1)*512 + m*2 + (k&1)
                v2f a = *reinterpret_cast<const v2f*>(
                    &panel[(2 * q + hi) * 512 + (mt * 16 + lm) * 2]);
                acc[mt] = __builtin_amdgcn_wmma_f32_16x16x4_f32(
                    false, a, false, bv[q], (short)0, acc[mt], false, false);
            }
        }
        p ^= 1;
    }

    // store C/D tiles: VGPR v -> M = mt*16 + v + 8*hi, N = Nbase + lm
#pragma unroll
    for (int mt = 0; mt < 16; ++mt) {
#pragma unroll
        for (int v = 0; v < 8; ++v) {
            const int M = mt * 16 + v + hi * 8;
            gates[((size_t)b * GATES + M) * HW + n] = acc[mt][v];
        }
    }
}

// ---------------------------------------------------------------------------
// LSTM gate nonlinearity + state update.
// grid = B*HID*HW/256 blocks x 256 threads
// ---------------------------------------------------------------------------
__global__ __launch_bounds__(256) void lstm_update_kernel(
    const float* __restrict__ gates, const float* __restrict__ bias,
    float* __restrict__ c_state, float* __restrict__ h_out,
    float* __restrict__ out1, int t) {
    const size_t idx = (size_t)blockIdx.x * 256 + threadIdx.x;  // < B*64*4096
    const int b  = (int)(idx >> 18);
    const int ch = (int)((idx >> 12) & 63);
    const int n  = (int)(idx & 4095);

    const size_t gb = ((size_t)b * GATES) * HW + n;
    float gi = gates[gb + (size_t)(ch)       * HW] + bias[ch];
    float gf = gates[gb + (size_t)(ch + 64)  * HW] + bias[ch + 64];
    float go = gates[gb + (size_t)(ch + 128) * HW] + bias[ch + 128];
    float gg = gates[gb + (size_t)(ch + 192) * HW] + bias[ch + 192];

    float c = sigmoidf(gf) * c_state[idx] + sigmoidf(gi) * tanhf(gg);
    float h = sigmoidf(go) * tanhf(c);
    c_state[idx] = c;
    h_out[idx] = h;
    if (out1)
        out1[(((size_t)b * TSTEPS + t) * HID + ch) * HW + n] = h;
}

// ---------------------------------------------------------------------------
extern "C" void kernel_launch(void* const* d_in, const int* in_sizes, int n_in,
                              void* d_out, int out_size, void* d_ws, size_t ws_size,
                              hipStream_t stream) {
    (void)in_sizes; (void)n_in; (void)out_size; (void)ws_size;
    const float* x   = (const float*)d_in[0];   // [8,16,3,64,64]
    const float* aw1 = (const float*)d_in[1];   // [1,3]
    const float* aw2 = (const float*)d_in[2];   // [3,1]
    const float* w0  = (const float*)d_in[3];   // [256,67,3,3]
    const float* b0  = (const float*)d_in[4];   // [256]
    const float* w1  = (const float*)d_in[5];   // [256,128,3,3]
    const float* b1  = (const float*)d_in[6];   // [256]

    float* ws  = (float*)d_ws;
    float* out = (float*)d_out;

    float* xin  = ws + XIN_OFF;
    float* h0   = ws + H0_OFF;
    float* c0   = ws + C0_OFF;
    float* gate = ws + GATE_OFF;
    float* Wt0  = ws + WT0_OFF;
    float* Wt1  = ws + WT1_OFF;
    float* h1   = out + H1_OFF;
    float* c1   = out + C1_OFF;

    // pack/pad weights into rs-major pair-interleaved layout; zero states
    wt_pack_kernel<67, 72><<<648, 256, 0, stream>>>(w0, Wt0);
    wt_pack_kernel<128, 128><<<1152, 256, 0, stream>>>(w1, Wt1);
    const size_t state_bytes = (size_t)BATCH * HID * HW * sizeof(float);
    (void)hipMemsetAsync(h0, 0, state_bytes, stream);
    (void)hipMemsetAsync(c0, 0, state_bytes, stream);
    (void)hipMemsetAsync(h1, 0, state_bytes, stream);
    (void)hipMemsetAsync(c1, 0, state_bytes, stream);

    attention_kernel<<<BATCH * TSTEPS, 256, 0, stream>>>(x, aw1, aw2, xin);

    const dim3 cgrid(32, BATCH);
    const int ugrid = (BATCH * HID * HW) / 256;
    for (int t = 0; t < TSTEPS; ++t) {
        // layer 0: input = [xin_t (3ch) | h0 (64ch)]
        conv_gates_kernel<3, 67, 72, 648><<<cgrid, 256, 0, stream>>>(
            xin + (size_t)t * 3 * HW, (long)TSTEPS * 3 * HW, h0, Wt0, gate);
        lstm_update_kernel<<<ugrid, 256, 0, stream>>>(gate, b0, c0, h0,
                                                      nullptr, 0);
        // layer 1: input = [h0 (64ch) | h1 (64ch)]
        conv_gates_kernel<HID, 128, 128, 1152><<<cgrid, 256, 0, stream>>>(
            h0, (long)HID * HW, h1, Wt1, gate);
        lstm_update_kernel<<<ugrid, 256, 0, stream>>>(gate, b1, c1, h1,
                                                      out + OUT1_OFF, t);
    }
}